// Supercomplex_55559696941650
// MI455X (gfx1250) — compile-verified
//
#include <hip/hip_runtime.h>
#include <hip/hip_bf16.h>
#include <stdint.h>

typedef float v2f __attribute__((ext_vector_type(2)));
typedef float v8f __attribute__((ext_vector_type(8)));

#define RANK 64
#define BATCH 256
#define N_ENT 400000
#define N_TS 365
#define NT 4                      // N-tiles (of 16) per block -> 64 columns/block

// ---------------------------------------------------------------------------
// Kernel 1: build G[256][64] = 2*(lhs*rel_ - te*tr) and bias[256]
// One block per batch row, 64 threads (one per k).
// ---------------------------------------------------------------------------
__global__ __launch_bounds__(64) void prep_kernel(
    const float* __restrict__ E0, const float* __restrict__ E1,
    const float* __restrict__ E2, const float* __restrict__ E3,
    const float* __restrict__ E4, const long long* __restrict__ x,
    float* __restrict__ G, float* __restrict__ bias) {
  const int i = blockIdx.x;      // batch row
  const int k = threadIdx.x;     // rank index 0..63

  const long long i0 = x[i * 4 + 0];
  const long long i1 = x[i * 4 + 1];
  const long long i3 = x[i * 4 + 3];

  const float lhs = E0[i0 * RANK + k];
  const float rel = E1[i1 * RANK + k];
  const float te  = E2[i3 * RANK + k];
  const float tr  = E3[i3 * RANK + k];
  const float ct  = E4[i3 * RANK + k];

  // complex_mul(rel, ct) with halves of size 32, then rel_ = rel + that.
  const int part = (k < 32) ? (k + 32) : (k - 32);
  const float relp = E1[i1 * RANK + part];
  const float ctp  = E4[i3 * RANK + part];

  float rel_;
  if (k < 32) {
    rel_ = rel + rel * ct + relp * ctp;        // a*c + b*d
  } else {
    rel_ = rel + relp * ct - rel * ctp;        // a*d - b*c
  }

  const float g = 2.0f * (lhs * rel_ - te * tr);
  G[i * RANK + k] = g;

  // bias contribution: 2*te*(te*rel_ + lhs*tr)
  __shared__ float red[RANK];
  red[k] = 2.0f * te * (te * rel_ + lhs * tr);
  __syncthreads();
  if (k == 0) {
    float s = 0.0f;
    #pragma unroll
    for (int t = 0; t < RANK; ++t) s += red[t];
    bias[i] = s;
  }
}

// ---------------------------------------------------------------------------
// Kernel 2: scores[m][n] = sum_k G[m][k] * E0[n][k] + bias[m]
// M=256, N=400000, K=64 via V_WMMA_F32_16X16X4_F32.
//
// Block: 512 threads = 16 waves; wave w owns M-tile [16w,16w+16) -> each block
// covers all 256 rows, so E0 streams exactly once device-wide.
// Block covers 64 N columns (4 accumulator tiles per wave): the 16 A
// fragments of G stay VGPR-resident and are reused 4x, so A traffic == output
// traffic instead of 4x it.
//
// Fragment layouts (ISA 7.12.2, 32-bit, wave32):
//   A (16x4):  lane L holds M=L%16; VGPR0 K=(L/16)*2, VGPR1 K=(L/16)*2+1
//   B (4x16):  lane L holds N=L%16; VGPR0 K=(L/16)*2, VGPR1 K=(L/16)*2+1
//   C/D f32:   VGPR r, lane L -> M = r + (L/16)*8, N = L%16
// ---------------------------------------------------------------------------
__global__ __launch_bounds__(512) void wmma_gemm_kernel(
    const float* __restrict__ E0, const float* __restrict__ G,
    const float* __restrict__ bias, float* __restrict__ scores) {
  const int lane = threadIdx.x & 31;
  const int wave = threadIdx.x >> 5;     // 0..15
  const int l16  = lane & 15;
  const int half = lane >> 4;            // 0 or 1 (K-pair select)
  const int m0   = wave * 16;
  const int n0   = blockIdx.x * (16 * NT);

  // Preload all 16 A fragments of the G tile (K = 0..63), VGPR-resident.
  v2f a[16];
  {
    const float* gp = G + (m0 + l16) * RANK + half * 2;
    #pragma unroll
    for (int s = 0; s < 16; ++s) a[s] = *(const v2f*)(gp + s * 4);
  }

  v8f acc[NT];
  #pragma unroll
  for (int t = 0; t < NT; ++t) acc[t] = (v8f){};

  #pragma unroll
  for (int t = 0; t < NT; ++t) {
    const float* bp = E0 + (size_t)(n0 + t * 16 + l16) * RANK + half * 2;
    #pragma unroll
    for (int s = 0; s < 16; ++s) {
      v2f b = *(const v2f*)(bp + s * 4);
      acc[t] = __builtin_amdgcn_wmma_f32_16x16x4_f32(
          /*neg_a=*/false, a[s], /*neg_b=*/false, b,
          /*c_mod=*/(short)0, acc[t], /*reuse_a=*/false, /*reuse_b=*/false);
    }
  }

  // Epilogue: add bias, store. Per row, each half-wave stores 64B contiguous
  // per tile; adjacent blocks fill adjacent 64-float segments (L2 combines).
  #pragma unroll
  for (int r = 0; r < 8; ++r) {
    const int m = m0 + r + half * 8;
    const float bv = bias[m];
    float* srow = scores + (size_t)m * N_ENT + n0 + l16;
    #pragma unroll
    for (int t = 0; t < NT; ++t) {
      srow[t * 16] = acc[t][r] + bv;
    }
  }
}

// ---------------------------------------------------------------------------
// Kernel 3: tail outputs: cl_loss = 0, then E2[:-1], E3[:-1], E4[:-1]
// ---------------------------------------------------------------------------
#define TAIL_COPY ((N_TS - 1) * RANK)   // 23296
__global__ void tail_kernel(const float* __restrict__ E2,
                            const float* __restrict__ E3,
                            const float* __restrict__ E4,
                            float* __restrict__ out_tail) {
  const int idx = blockIdx.x * blockDim.x + threadIdx.x;
  const int total = 1 + 3 * TAIL_COPY;
  if (idx >= total) return;
  if (idx == 0) { out_tail[0] = 0.0f; return; }
  int j = idx - 1;
  float v;
  if (j < TAIL_COPY)            v = E2[j];
  else if (j < 2 * TAIL_COPY)   v = E3[j - TAIL_COPY];
  else                          v = E4[j - 2 * TAIL_COPY];
  out_tail[idx] = v;
}

// ---------------------------------------------------------------------------
extern "C" void kernel_launch(void* const* d_in, const int* in_sizes, int n_in,
                              void* d_out, int out_size, void* d_ws, size_t ws_size,
                              hipStream_t stream) {
  const float* E0 = (const float*)d_in[0];
  const float* E1 = (const float*)d_in[1];
  const float* E2 = (const float*)d_in[2];
  const float* E3 = (const float*)d_in[3];
  const float* E4 = (const float*)d_in[4];
  const long long* x = (const long long*)d_in[5];

  float* scores = (float*)d_out;                       // 256 * 400000
  float* out_tail = scores + (size_t)BATCH * N_ENT;    // 1 + 3*23296

  float* G    = (float*)d_ws;                          // 256*64 floats
  float* bias = G + BATCH * RANK;                      // 256 floats

  prep_kernel<<<BATCH, RANK, 0, stream>>>(E0, E1, E2, E3, E4, x, G, bias);

  wmma_gemm_kernel<<<N_ENT / (16 * NT), 512, 0, stream>>>(E0, G, bias, scores);

  const int tail_total = 1 + 3 * TAIL_COPY;
  tail_kernel<<<(tail_total + 255) / 256, 256, 0, stream>>>(E2, E3, E4, out_tail);
}